// GraphConvNet_77249281786532
// MI455X (gfx1250) — compile-verified
//
#include <hip/hip_runtime.h>

// MI455X flash-attention formulation of:
//   support = x^T W ; adj = softmax(x^T x) ; out = (adj @ support)^T
// All matrix math via V_WMMA_F32_16X16X4_F32 (full fp32 matrix pipe).
// Traffic kept to ~16MB (L2-resident); compute-bound on WMMA.

#define BATCH   4
#define FDIM    64
#define N_NODES 8192
#define LROW    68      // 64 + 4 pad floats: bank-conflict-free scattered reads

typedef __attribute__((ext_vector_type(2))) float v2f;
typedef __attribute__((ext_vector_type(8))) float v8f;

// ---------------------------------------------------------------------------
// Prep: xt[b][n][f] = x[b][f][n];  sup[b][n][o] = sum_f xt[b][n][f] * W[f][o]
// ---------------------------------------------------------------------------
__global__ __launch_bounds__(256) void gcn_prep(const float* __restrict__ x,
                                                const float* __restrict__ w,
                                                float* __restrict__ xt,
                                                float* __restrict__ sup) {
    int n = blockIdx.x * 256 + threadIdx.x;   // node
    int b = blockIdx.y;                        // batch
    const float* xb = x + (size_t)b * FDIM * N_NODES;
    float* xtrow = xt  + ((size_t)b * N_NODES + n) * FDIM;
    float* srow  = sup + ((size_t)b * N_NODES + n) * FDIM;

    float acc[FDIM];
#pragma unroll
    for (int o = 0; o < FDIM; ++o) acc[o] = 0.0f;

    for (int f = 0; f < FDIM; ++f) {
        float v = xb[(size_t)f * N_NODES + n];   // coalesced across lanes
        xtrow[f] = v;
#pragma unroll
        for (int o = 0; o < FDIM; ++o) acc[o] += v * w[f * FDIM + o]; // w: s_loads
    }
#pragma unroll
    for (int o = 0; o < FDIM; ++o) srow[o] = acc[o];
}

// ---------------------------------------------------------------------------
// Flash kernel: block = 128 threads (4 waves). Block covers 64 query rows,
// each wave owns one 16-row query tile. K/V tiles (16 rows x 64 f32 each)
// staged in double-buffered LDS by all 4 waves, global loads overlapped
// with WMMA compute.
// ---------------------------------------------------------------------------
__device__ __forceinline__ void stage_load(const float* __restrict__ xtb,
                                           const float* __restrict__ vbp,
                                           int j, int tid,
                                           float4* rk, float4* rv) {
#pragma unroll
    for (int i = 0; i < 2; ++i) {
        int idx = i * 128 + tid;          // 0..255 float4 slots of the tile
        int row = idx >> 4;               // 0..15
        int c4  = idx & 15;               // 0..15 (x4 floats)
        size_t off = (size_t)(j * 16 + row) * FDIM + c4 * 4;
        rk[i] = *(const float4*)(xtb + off);
        rv[i] = *(const float4*)(vbp + off);
    }
}

__global__ __launch_bounds__(128) void gcn_flash(const float* __restrict__ xt,
                                                 const float* __restrict__ sup,
                                                 float* __restrict__ out) {
    __shared__ __align__(16) float sh[2][2][16][LROW];  // [buf][k=0/v=1][row][col]

    const int tid  = threadIdx.x;
    const int wave = tid >> 5;
    const int lane = tid & 31;
    const int hi   = lane >> 4;     // which 16-lane half
    const int mrow = lane & 15;     // row index within 16 (M for A, N for B)

    const int b    = blockIdx.x >> 7;        // 128 q-blocks per batch
    const int qblk = blockIdx.x & 127;
    const int qbase = qblk * 64 + wave * 16;

    const float* xtb = xt  + (size_t)b * N_NODES * FDIM;
    const float* vbp = sup + (size_t)b * N_NODES * FDIM;

    // Preload Q in B-operand layout: chunk c -> features {4c+2*hi, 4c+2*hi+1}
    // of query row (qbase + mrow).  (8-byte aligned -> global_load_b64)
    v2f q[16];
    {
        const float* qrow = xtb + (size_t)(qbase + mrow) * FDIM + 2 * hi;
#pragma unroll
        for (int c = 0; c < 16; ++c)
            q[c] = *(const v2f*)(qrow + 4 * c);
    }

    v8f accT[4];                      // O^T, four 16x16 f32 tiles (o-major)
#pragma unroll
    for (int t = 0; t < 4; ++t)
#pragma unroll
        for (int r = 0; r < 8; ++r) accT[t][r] = 0.0f;

    float m = -3.0e38f;               // running row max
    float l = 0.0f;                   // running row sum

    float4 rk[2], rv[2];
    stage_load(xtb, vbp, 0, tid, rk, rv);

    const int NJ = N_NODES / 16;      // 512 key tiles
    for (int j = 0; j < NJ; ++j) {
        const int buf = j & 1;
        // write staged tile to LDS
#pragma unroll
        for (int i = 0; i < 2; ++i) {
            int idx = i * 128 + tid;
            int row = idx >> 4, c4 = idx & 15;
            *(float4*)&sh[buf][0][row][c4 * 4] = rk[i];
            *(float4*)&sh[buf][1][row][c4 * 4] = rv[i];
        }
        __syncthreads();
        if (j + 1 < NJ)               // overlap next tile's global loads
            stage_load(xtb, vbp, j + 1, tid, rk, rv);

        const float* kb    = &sh[buf][0][0][0];
        const float* vtile = &sh[buf][1][0][0];

        // ---- S^T = K_tile * Q^T  (16 chained fp32 WMMAs over feature dim)
        v8f s;
#pragma unroll
        for (int r = 0; r < 8; ++r) s[r] = 0.0f;
#pragma unroll
        for (int c = 0; c < 16; ++c) {
            v2f kk = *(const v2f*)(kb + mrow * LROW + 4 * c + 2 * hi);
            s = __builtin_amdgcn_wmma_f32_16x16x4_f32(
                    false, kk, false, q[c], (short)0, s, false, false);
        }

        // ---- streaming softmax: lane L owns query row L%16 (8 cols here,
        //      mirror lane L^16 has the other 8) -> one xor-16 shuffle.
        float mloc = s[0];
#pragma unroll
        for (int r = 1; r < 8; ++r) mloc = fmaxf(mloc, s[r]);
        mloc = fmaxf(mloc, __shfl_xor(mloc, 16, 32));
        float mnew = fmaxf(m, mloc);
        float sc = __expf(m - mnew);
        m = mnew;

        float p[8], ls = 0.0f;
#pragma unroll
        for (int r = 0; r < 8; ++r) { p[r] = __expf(s[r] - mnew); ls += p[r]; }
        ls += __shfl_xor(ls, 16, 32);
        l = l * sc + ls;

#pragma unroll
        for (int t = 0; t < 4; ++t)
#pragma unroll
            for (int r = 0; r < 8; ++r) accT[t][r] *= sc;  // per-lane scalar

        // ---- cross-half exchange builds P^T directly in B-operand layout
        float tt[8];
#pragma unroll
        for (int r = 0; r < 8; ++r) tt[r] = __shfl_xor(p[r], 16, 32);
        v2f pb[4];
        pb[0].x = hi ? tt[2] : p[0];  pb[0].y = hi ? tt[3] : p[1];  // k0=0
        pb[1].x = hi ? tt[6] : p[4];  pb[1].y = hi ? tt[7] : p[5];  // k0=4
        pb[2].x = hi ? p[2] : tt[0];  pb[2].y = hi ? p[3] : tt[1];  // k0=8
        pb[3].x = hi ? p[6] : tt[4];  pb[3].y = hi ? p[7] : tt[5];  // k0=12

        // ---- O^T += V^T * P^T  (4 o-tiles x 4 k-chunks of fp32 WMMA)
#pragma unroll
        for (int t = 0; t < 4; ++t) {
#pragma unroll
            for (int c = 0; c < 4; ++c) {
                int krow = 4 * c + 2 * hi;
                v2f a;
                a.x = vtile[krow * LROW + 16 * t + mrow];
                a.y = vtile[(krow + 1) * LROW + 16 * t + mrow];
                accT[t] = __builtin_amdgcn_wmma_f32_16x16x4_f32(
                              false, a, false, pb[c], (short)0, accT[t],
                              false, false);
            }
        }
    }

    // ---- epilogue: divide by row sum, store out[b][o][n] (coalesced in n)
    float inv = 1.0f / l;
    int n = qbase + mrow;
#pragma unroll
    for (int t = 0; t < 4; ++t)
#pragma unroll
        for (int r = 0; r < 8; ++r) {
            int o = 16 * t + r + 8 * hi;
            out[((size_t)b * FDIM + o) * N_NODES + n] = accT[t][r] * inv;
        }
}

// ---------------------------------------------------------------------------
extern "C" void kernel_launch(void* const* d_in, const int* in_sizes, int n_in,
                              void* d_out, int out_size, void* d_ws, size_t ws_size,
                              hipStream_t stream) {
    (void)in_sizes; (void)n_in; (void)out_size; (void)ws_size;
    const float* x = (const float*)d_in[0];     // [4,64,8192] f32
    const float* w = (const float*)d_in[1];     // [64,64] f32
    float* xt  = (float*)d_ws;                                   // 8 MB
    float* sup = xt + (size_t)BATCH * N_NODES * FDIM;            // 8 MB
    float* out = (float*)d_out;                                  // [4,64,8192]

    dim3 gPrep(N_NODES / 256, BATCH);
    gcn_prep<<<gPrep, 256, 0, stream>>>(x, w, xt, sup);

    dim3 gFlash(BATCH * (N_NODES / 64));
    gcn_flash<<<gFlash, 128, 0, stream>>>(xt, sup, out);
}